// NetTrans_64398739636346
// MI455X (gfx1250) — compile-verified
//
#include <hip/hip_runtime.h>

typedef __attribute__((ext_vector_type(2))) float v2f;
typedef __attribute__((ext_vector_type(8))) float v8f;

#define N1n   50000
#define N2n   50000
#define E1n   800000
#define E2n   800000
#define An    10000
#define NSUP  25000
#define Dn    128
#define KHOPS 10

// d_out element offsets (fp32), in reference return order
#define OUTX_OFF  0L
#define SUP1_OFF  6400000L
#define OUTY_OFF  9600000L
#define SUP2_OFF  16000000L
#define AVEC_OFF  19200000L
#define RECON_OFF 19250000L

__device__ __forceinline__ unsigned fkey(float f) {
  unsigned u = __float_as_uint(f);
  return (u & 0x80000000u) ? ~u : (u | 0x80000000u);
}
__device__ __forceinline__ float finv(unsigned k) {
  unsigned u = (k & 0x80000000u) ? (k ^ 0x80000000u) : ~k;
  return __uint_as_float(u);
}

// ---------------- utility fills ----------------
__global__ void fill_f32(float* __restrict__ p, float v, long n) {
  long t = (long)blockIdx.x * blockDim.x + threadIdx.x;
  if (t < n) p[t] = v;
}
__global__ void fill_i32(int* __restrict__ p, int v, long n) {
  long t = (long)blockIdx.x * blockDim.x + threadIdx.x;
  if (t < n) p[t] = v;
}

// ---------------- degree / edge norm ----------------
__global__ void deg_kernel(const int* __restrict__ dst, const float* __restrict__ w,
                           float* __restrict__ deg, int E) {
  long t = (long)blockIdx.x * blockDim.x + threadIdx.x;
  if (t < E) atomicAdd(deg + dst[t], w[t]);
}
__global__ void norm_kernel(const int* __restrict__ src, const int* __restrict__ dst,
                            const float* __restrict__ w, const float* __restrict__ deg,
                            float* __restrict__ norm, int E) {
  long t = (long)blockIdx.x * blockDim.x + threadIdx.x;
  if (t >= E) return;
  float ds = deg[src[t]] + 1.0f, dd = deg[dst[t]] + 1.0f;
  norm[t] = w[t] * rsqrtf(ds * dd);
}

// ---------------- weighted message passing: out[dst] += norm * h[src] ----------------
__global__ void prop_kernel(const float* __restrict__ h, const int* __restrict__ src,
                            const int* __restrict__ dst, const float* __restrict__ norm,
                            float* __restrict__ out, int E) {
  long t = (long)blockIdx.x * blockDim.x + threadIdx.x;
  int e = (int)(t >> 5);
  if (e >= E) return;
  int f = ((int)t & 31) << 2;
  float n = norm[e];
  const float4 v = *(const float4*)(h + (long)src[e] * Dn + f);
  float* o = out + (long)dst[e] * Dn + f;
  atomicAdd(o + 0, n * v.x);
  atomicAdd(o + 1, n * v.y);
  atomicAdd(o + 2, n * v.z);
  atomicAdd(o + 3, n * v.w);
}

// ---------------- fp32 WMMA GEMM: Y[nrows x 128] = gather(X) @ W + bias ----------------
// one wave -> one 16x16 tile of Y; block = 256 threads = 8 waves = 16 rows x 128 cols.
// A tile (16x128) is async-staged once per block into LDS (padded to stride 132 to keep
// the 16 row-readers per half-wave on distinct banks); W panel stays in L2/WGP$ and is
// prefetched. K loop fully unrolled: 32 chained v_wmma_f32_16x16x4_f32 per wave.
__global__ void gemm128_wmma(const float* __restrict__ X, const int* __restrict__ gather,
                             const float* __restrict__ W, const float* __restrict__ bias,
                             float* __restrict__ Y, int nrows, int relu) {
  __shared__ float Alds[16 * 132];                 // 8448 B
  const int tid  = threadIdx.x;
  const int row0 = blockIdx.x << 4;

  // pull the 64KB weight panel toward this WGP (global_prefetch_b8)
  __builtin_prefetch(W + tid * 32, 0, 2);
  __builtin_prefetch(W + 8192 + tid * 32, 0, 2);

  // async-stage A tile: 512 float4 chunks, 2 per thread (ASYNCcnt path)
  unsigned lbase = (unsigned)(size_t)(void*)Alds;
  for (int it = 0; it < 2; ++it) {
    int c  = tid + (it << 8);
    int r  = c >> 5;                               // row within tile
    int cc = (c & 31) << 2;                        // starting float within row
    int row = row0 + r;
    int rcl = row < nrows ? row : nrows - 1;       // clamp: EXEC stays all-ones
    long srow = gather ? (long)gather[rcl] : (long)rcl;
    unsigned goff  = (unsigned)((srow * Dn + cc) * 4);
    unsigned laddr = lbase + (unsigned)((r * 132 + cc) * 4);
    asm volatile("global_load_async_to_lds_b128 %0, %1, %2"
                 :: "v"(laddr), "v"(goff), "s"(X)
                 : "memory");
  }
  asm volatile("s_wait_asynccnt 0x0" ::: "memory");
  __syncthreads();

  const int lane = tid & 31;
  const int l16  = lane & 15;
  const int hi   = lane >> 4;                      // 0: K pair {0,1}; 1: K pair {2,3}
  const int col  = ((tid >> 5) << 4) | l16;
  const float* ar = &Alds[l16 * 132];

  v8f acc = {};
#pragma unroll
  for (int k0 = 0; k0 < Dn; k0 += 4) {
    int ka = k0 + (hi << 1);
    v2f a, b;
    a.x = ar[ka];
    a.y = ar[ka + 1];
    b.x = W[ka * Dn + col];
    b.y = W[(ka + 1) * Dn + col];
    acc = __builtin_amdgcn_wmma_f32_16x16x4_f32(false, a, false, b, (short)0, acc,
                                                false, false);
  }
  float bv = bias ? bias[col] : 0.0f;
#pragma unroll
  for (int r = 0; r < 8; ++r) {
    int orow = row0 + r + (hi << 3);
    if (orow < nrows) {
      float v = acc[r] + bv;
      if (relu) v = fmaxf(v, 0.0f);
      Y[(long)orow * Dn + col] = v;
    }
  }
}

// ---------------- s[i] = dot(relu(outx[i]), p) ----------------
__global__ void score_kernel(const float* __restrict__ outx, const float* __restrict__ p,
                             float* __restrict__ s, int N) {
  long t = (long)blockIdx.x * blockDim.x + threadIdx.x;
  int i = (int)(t >> 5);
  if (i >= N) return;
  int f = ((int)t & 31) << 2;
  const float4 v = *(const float4*)(outx + (long)i * Dn + f);
  const float4 q = *(const float4*)(p + f);
  float acc = fmaxf(v.x, 0.f) * q.x + fmaxf(v.y, 0.f) * q.y +
              fmaxf(v.z, 0.f) * q.z + fmaxf(v.w, 0.f) * q.w;
  for (int m = 16; m >= 1; m >>= 1) acc += __shfl_xor(acc, m, 32);
  if (((int)t & 31) == 0) s[i] = acc;
}

// ---------------- exact top-k via rank (jax.lax.top_k tie-break: lower index first) ----------------
__global__ void rank_kernel(const float* __restrict__ s, int* __restrict__ supid,
                            int n, int nsup) {
  __shared__ float tile[256];
  int i = blockIdx.x * 256 + threadIdx.x;
  float si = (i < n) ? s[i] : 0.0f;
  int rank = 0;
  for (int base = 0; base < n; base += 256) {
    int j = base + threadIdx.x;
    tile[threadIdx.x] = (j < n) ? s[j] : -3.0e38f;
    __syncthreads();
    int lim = min(256, n - base);
    for (int u = 0; u < lim; ++u) {
      float sj = tile[u];
      int jj = base + u;
      rank += (int)((sj > si) || (sj == si && jj < i));
    }
    __syncthreads();
  }
  if (i < n) supid[i] = (rank < nsup) ? rank : -1;
}

// ---------------- attention passes over candidates (E edges + N self-loops) ----------------
__global__ void attA_kernel(const float* __restrict__ outx, const int* __restrict__ src,
                            const int* __restrict__ dst, const int* __restrict__ supid,
                            const int* __restrict__ temp_raw, float* __restrict__ att,
                            unsigned* __restrict__ mkey, int E, int N) {
  long t = (long)blockIdx.x * blockDim.x + threadIdx.x;
  long c = t >> 5;
  long C = (long)E + N;
  if (c >= C) return;
  int lane = (int)t & 31;
  int node = (c < E) ? src[c] : (int)(c - E);
  int sup  = (c < E) ? dst[c] : (int)(c - E);
  int f = lane << 2;
  const float4 a4 = *(const float4*)(outx + (long)node * Dn + f);
  const float4 b4 = *(const float4*)(outx + (long)sup * Dn + f);
  float acc = fmaxf(a4.x, 0.f) * fmaxf(b4.x, 0.f) + fmaxf(a4.y, 0.f) * fmaxf(b4.y, 0.f) +
              fmaxf(a4.z, 0.f) * fmaxf(b4.z, 0.f) + fmaxf(a4.w, 0.f) * fmaxf(b4.w, 0.f);
  for (int m = 16; m >= 1; m >>= 1) acc += __shfl_xor(acc, m, 32);
  if (lane == 0) {
    int tv = temp_raw[0];  // scalar: int 1 or float bits
    float T = (tv >= 1 && tv < 1000) ? (float)tv : __int_as_float(tv);
    float v = acc / (sqrtf(128.0f) * T);
    bool valid = supid[sup] >= 0;
    v = valid ? v : -1.0e30f;
    att[c] = v;
    atomicMax(mkey + node, fkey(v));
  }
}

__global__ void attB_kernel(const float* __restrict__ att, const int* __restrict__ src,
                            const unsigned* __restrict__ mkey, float* __restrict__ esum,
                            int E, int N) {
  long c = (long)blockIdx.x * blockDim.x + threadIdx.x;
  long C = (long)E + N;
  if (c >= C) return;
  int node = (c < E) ? src[c] : (int)(c - E);
  float e = expf(att[c] - finv(mkey[node]));
  atomicAdd(esum + node, e);
}

__global__ void attC_kernel(const float* __restrict__ att, const int* __restrict__ src,
                            const int* __restrict__ dst, const int* __restrict__ supid,
                            const unsigned* __restrict__ mkey, int* __restrict__ cluster,
                            int E, int N) {
  long c = (long)blockIdx.x * blockDim.x + threadIdx.x;
  long C = (long)E + N;
  if (c >= C) return;
  int node = (c < E) ? src[c] : (int)(c - E);
  int sup  = (c < E) ? dst[c] : (int)(c - E);
  int sid = supid[sup];
  if (sid >= 0 && att[c] >= finv(mkey[node])) atomicMax(cluster + node, sid);
}

__global__ void attD_kernel(const float* __restrict__ att, const int* __restrict__ src,
                            const int* __restrict__ dst, const int* __restrict__ supid,
                            const unsigned* __restrict__ mkey, const float* __restrict__ esum,
                            const int* __restrict__ cluster, int* __restrict__ abits,
                            int E, int N) {
  long c = (long)blockIdx.x * blockDim.x + threadIdx.x;
  long C = (long)E + N;
  if (c >= C) return;
  int node = (c < E) ? src[c] : (int)(c - E);
  int sup  = (c < E) ? dst[c] : (int)(c - E);
  int sid = supid[sup];
  float m = finv(mkey[node]);
  if (sid >= 0 && att[c] >= m && sid == cluster[node]) {
    float w = expf(att[c] - m) / esum[node];     // w >= 0 -> raw bits monotone
    atomicMax(abits + node, __float_as_int(w));
  }
}

__global__ void finalize_kernel(int* __restrict__ cluster, const int* __restrict__ abits,
                                float* __restrict__ a_out, int N) {
  long i = (long)blockIdx.x * blockDim.x + threadIdx.x;
  if (i >= N) return;
  int cl = cluster[i];
  bool has = cl >= 0;
  cluster[i] = has ? cl : 0;
  a_out[i] = has ? __int_as_float(abits[i]) : 0.0f;
}

// ---------------- coarsening ----------------
__global__ void pool_kernel(const float* __restrict__ outx, const int* __restrict__ cluster,
                            const float* __restrict__ a, float* __restrict__ xpool, int N) {
  long t = (long)blockIdx.x * blockDim.x + threadIdx.x;
  int i = (int)(t >> 5);
  if (i >= N) return;
  float av = a[i];
  if (av == 0.0f) return;
  int f = ((int)t & 31) << 2;
  const float4 v = *(const float4*)(outx + (long)i * Dn + f);
  float* o = xpool + (long)cluster[i] * Dn + f;
  atomicAdd(o + 0, av * fmaxf(v.x, 0.f));
  atomicAdd(o + 1, av * fmaxf(v.y, 0.f));
  atomicAdd(o + 2, av * fmaxf(v.z, 0.f));
  atomicAdd(o + 3, av * fmaxf(v.w, 0.f));
}

__global__ void coarse_map_kernel(const int* __restrict__ src, const int* __restrict__ dst,
                                  const float* __restrict__ w, const int* __restrict__ cluster,
                                  const float* __restrict__ a, int* __restrict__ csrc,
                                  int* __restrict__ cdst, float* __restrict__ cw, int E) {
  long e = (long)blockIdx.x * blockDim.x + threadIdx.x;
  if (e >= E) return;
  int s = src[e], d = dst[e];
  csrc[e] = cluster[s];
  cdst[e] = cluster[d];
  cw[e] = w[e] * a[s] * a[d];
}

__global__ void add_relu_kernel(const float* __restrict__ pa, const float* __restrict__ pb,
                                float* __restrict__ out, long n) {
  long t = (long)blockIdx.x * blockDim.x + threadIdx.x;
  if (t < n) out[t] = fmaxf(pa[t] + pb[t], 0.0f);
}

// ---------------- anchors ----------------
__global__ void g1_kernel(const int* __restrict__ a1, const int* __restrict__ cluster,
                          int* __restrict__ g1, int A) {
  long k = (long)blockIdx.x * blockDim.x + threadIdx.x;
  if (k < A) g1[k] = cluster[a1[k]];
}

__global__ void anchor_scatter_kernel(const int* __restrict__ a1, const int* __restrict__ a2,
                                      const float* __restrict__ a, const float* __restrict__ t1g,
                                      const float* __restrict__ t2g, float* __restrict__ outy,
                                      int A) {
  long t = (long)blockIdx.x * blockDim.x + threadIdx.x;
  int k = (int)(t >> 5);
  if (k >= A) return;
  int f = ((int)t & 31) << 2;
  float qw = a[a1[k]];
  const float4 u = *(const float4*)(t1g + (long)k * Dn + f);
  const float4 v = *(const float4*)(t2g + (long)k * Dn + f);
  float* o = outy + (long)a2[k] * Dn + f;
  atomicAdd(o + 0, qw * u.x + v.x);
  atomicAdd(o + 1, qw * u.y + v.y);
  atomicAdd(o + 2, qw * u.z + v.z);
  atomicAdd(o + 3, qw * u.w + v.w);
}

// ============================ host ============================
static inline unsigned gblk(long n, int b) { return (unsigned)((n + b - 1) / b); }

extern "C" void kernel_launch(void* const* d_in, const int* in_sizes, int n_in,
                              void* d_out, int out_size, void* d_ws, size_t ws_size,
                              hipStream_t stream) {
  (void)in_sizes; (void)n_in; (void)out_size; (void)ws_size;
  const float* x      = (const float*)d_in[0];
  const int*   src1   = (const int*)d_in[1];
  const int*   dst1   = src1 + E1n;
  const float* w1     = (const float*)d_in[2];
  const float* y      = (const float*)d_in[3];
  const int*   srcy   = (const int*)d_in[4];
  const int*   dsty   = srcy + E2n;
  const float* wy     = (const float*)d_in[5];
  const int*   a1     = (const int*)d_in[6];
  const int*   a2     = a1 + An;
  const int*   temp   = (const int*)d_in[7];
  const float* Wgcn   = (const float*)d_in[8];
  const float* bgcn   = (const float*)d_in[9];
  const float* pscore = (const float*)d_in[10];
  const float* Wc1    = (const float*)d_in[11];
  const float* bc1    = (const float*)d_in[12];
  const float* Wm1    = (const float*)d_in[13];
  const float* bm1    = (const float*)d_in[14];
  const float* Wl1    = (const float*)d_in[15];
  const float* Wl2    = (const float*)d_in[16];
  const float* Wm2    = (const float*)d_in[17];
  const float* bm2    = (const float*)d_in[18];

  float* out  = (float*)d_out;
  float* OUTX = out + OUTX_OFF;
  float* SUP1 = out + SUP1_OFF;
  float* OUTY = out + OUTY_OFF;
  float* SUP2 = out + SUP2_OFF;
  float* AVEC = out + AVEC_OFF;
  float* RECON = out + RECON_OFF;

  // bump allocator over workspace
  char* wsb = (char*)d_ws;
  size_t off = 0;
  auto alloc = [&](size_t bytes) -> void* {
    off = (off + 255) & ~(size_t)255;
    void* p = wsb + off;
    off += bytes;
    return p;
  };
  float*    hA    = (float*)alloc((size_t)N1n * Dn * 4);
  float*    hB    = (float*)alloc((size_t)N1n * Dn * 4);
  float*    normb = (float*)alloc((size_t)E1n * 4);       // norm1 -> norm_c -> normy
  float*    deg   = (float*)alloc((size_t)N1n * 4);
  float*    sarr  = (float*)alloc((size_t)N1n * 4);
  int*      supid = (int*)alloc((size_t)N1n * 4);
  float*    att   = (float*)alloc((size_t)(E1n + N1n) * 4);
  unsigned* mkey  = (unsigned*)alloc((size_t)N1n * 4);
  float*    esum  = (float*)alloc((size_t)N1n * 4);
  int*      clus  = (int*)alloc((size_t)N1n * 4);
  int*      abits = (int*)alloc((size_t)N1n * 4);
  float*    xpool = (float*)alloc((size_t)NSUP * Dn * 4);
  float*    cw    = (float*)alloc((size_t)E1n * 4);
  int*      csrc  = (int*)alloc((size_t)E1n * 4);
  int*      cdst  = (int*)alloc((size_t)E1n * 4);
  float*    t1g   = (float*)alloc((size_t)An * Dn * 4);
  float*    t2g   = (float*)alloc((size_t)An * Dn * 4);
  int*      g1    = (int*)alloc((size_t)An * 4);

  const int B = 256;
  long Cc = (long)E1n + N1n;  // candidate count

  // ---- graph 1 norm ----
  fill_f32<<<gblk(N1n, B), B, 0, stream>>>(deg, 0.0f, N1n);
  deg_kernel<<<gblk(E1n, B), B, 0, stream>>>(dst1, w1, deg, E1n);
  norm_kernel<<<gblk(E1n, B), B, 0, stream>>>(src1, dst1, w1, deg, normb, E1n);

  // ---- 10-hop GCN on graph 1 -> OUTX ----
  gemm128_wmma<<<gblk(N1n, 16), B, 0, stream>>>(x, nullptr, Wgcn, bgcn, hA, N1n, 0);
  {
    float* cur = hA;
    for (int k = 0; k < KHOPS; ++k) {
      float* nxt = (k == KHOPS - 1) ? OUTX : (cur == hA ? hB : hA);
      fill_f32<<<gblk((long)N1n * Dn, B), B, 0, stream>>>(nxt, 0.0f, (long)N1n * Dn);
      prop_kernel<<<gblk((long)E1n * 32, B), B, 0, stream>>>(cur, src1, dst1, normb, nxt, E1n);
      cur = nxt;
    }
  }

  // ---- TransPool: scores, exact top-k ranks ----
  score_kernel<<<gblk((long)N1n * 32, B), B, 0, stream>>>(OUTX, pscore, sarr, N1n);
  rank_kernel<<<gblk(N1n, B), B, 0, stream>>>(sarr, supid, N1n, NSUP);

  // ---- attention assignment (4 passes) ----
  fill_i32<<<gblk(N1n, B), B, 0, stream>>>((int*)mkey, 0, N1n);
  attA_kernel<<<gblk(Cc * 32, B), B, 0, stream>>>(OUTX, src1, dst1, supid, temp, att, mkey,
                                                  E1n, N1n);
  fill_f32<<<gblk(N1n, B), B, 0, stream>>>(esum, 0.0f, N1n);
  attB_kernel<<<gblk(Cc, B), B, 0, stream>>>(att, src1, mkey, esum, E1n, N1n);
  fill_i32<<<gblk(N1n, B), B, 0, stream>>>(clus, -1, N1n);
  attC_kernel<<<gblk(Cc, B), B, 0, stream>>>(att, src1, dst1, supid, mkey, clus, E1n, N1n);
  fill_i32<<<gblk(N1n, B), B, 0, stream>>>(abits, 0, N1n);
  attD_kernel<<<gblk(Cc, B), B, 0, stream>>>(att, src1, dst1, supid, mkey, esum, clus, abits,
                                             E1n, N1n);
  finalize_kernel<<<gblk(N1n, B), B, 0, stream>>>(clus, abits, AVEC, N1n);

  // ---- coarsen ----
  fill_f32<<<gblk((long)NSUP * Dn, B), B, 0, stream>>>(xpool, 0.0f, (long)NSUP * Dn);
  pool_kernel<<<gblk((long)N1n * 32, B), B, 0, stream>>>(OUTX, clus, AVEC, xpool, N1n);
  coarse_map_kernel<<<gblk(E1n, B), B, 0, stream>>>(src1, dst1, w1, clus, AVEC, csrc, cdst, cw,
                                                    E1n);
  fill_f32<<<gblk(NSUP, B), B, 0, stream>>>(deg, 0.0f, NSUP);
  deg_kernel<<<gblk(E1n, B), B, 0, stream>>>(cdst, cw, deg, E1n);
  norm_kernel<<<gblk(E1n, B), B, 0, stream>>>(csrc, cdst, cw, deg, normb, E1n);

  // ---- coarse GCNConv with self loop -> SUP1 ----
  gemm128_wmma<<<gblk(NSUP, 16), B, 0, stream>>>(xpool, nullptr, Wc1, bc1, hA, NSUP, 0);
  fill_f32<<<gblk((long)NSUP * Dn, B), B, 0, stream>>>(hB, 0.0f, (long)NSUP * Dn);
  prop_kernel<<<gblk((long)E1n * 32, B), B, 0, stream>>>(hA, csrc, cdst, normb, hB, E1n);
  add_relu_kernel<<<gblk((long)NSUP * Dn, B), B, 0, stream>>>(hA, hB, SUP1, (long)NSUP * Dn);

  // ---- MLP1 -> SUP2 ----
  gemm128_wmma<<<gblk(NSUP, 16), B, 0, stream>>>(SUP1, nullptr, Wm1, bm1, SUP2, NSUP, 1);

  // ---- graph 2: shared 10-hop GCN -> OUTY ----
  fill_f32<<<gblk(N2n, B), B, 0, stream>>>(deg, 0.0f, N2n);
  deg_kernel<<<gblk(E2n, B), B, 0, stream>>>(dsty, wy, deg, E2n);
  norm_kernel<<<gblk(E2n, B), B, 0, stream>>>(srcy, dsty, wy, deg, normb, E2n);
  gemm128_wmma<<<gblk(N2n, 16), B, 0, stream>>>(y, nullptr, Wgcn, bgcn, hA, N2n, 0);
  {
    float* cur = hA;
    for (int k = 0; k < KHOPS; ++k) {
      float* nxt = (k == KHOPS - 1) ? OUTY : (cur == hA ? hB : hA);
      fill_f32<<<gblk((long)N2n * Dn, B), B, 0, stream>>>(nxt, 0.0f, (long)N2n * Dn);
      prop_kernel<<<gblk((long)E2n * 32, B), B, 0, stream>>>(cur, srcy, dsty, normb, nxt, E2n);
      cur = nxt;
    }
  }

  // ---- LastLayer anchor scatters into OUTY ----
  g1_kernel<<<gblk(An, B), B, 0, stream>>>(a1, clus, g1, An);
  gemm128_wmma<<<gblk(An, 16), B, 0, stream>>>(SUP2, g1, Wl1, nullptr, t1g, An, 0);
  gemm128_wmma<<<gblk(An, 16), B, 0, stream>>>(OUTX, a1, Wl2, nullptr, t2g, An, 0);
  anchor_scatter_kernel<<<gblk((long)An * 32, B), B, 0, stream>>>(a1, a2, AVEC, t1g, t2g, OUTY,
                                                                  An);

  // ---- recon_y = OUTY @ W_mlp2 + b ----
  gemm128_wmma<<<gblk(N2n, 16), B, 0, stream>>>(OUTY, nullptr, Wm2, bm2, RECON, N2n, 0);
}